// GAT_90795608637583
// MI455X (gfx1250) — compile-verified
//
#include <hip/hip_runtime.h>
#include <math.h>

#define NODES 50000

typedef float v2f __attribute__((ext_vector_type(2)));
typedef float v8f __attribute__((ext_vector_type(8)));

// ---------------------------------------------------------------------------
// C = A(MxK) * B(K x NCOLS), all f32, via V_WMMA_F32_16X16X4_F32.
// One wave32 per 16x16 output tile; K consumed 4 at a time.
// A frag (16x4): lane L holds row M = L&15, K = khalf + {0,1}, khalf = 2*(L>>4)
// B frag (4x16): lane L holds col N = L&15, rows khalf + {0,1}
// C/D (16x16):  VGPR r, lane L -> row 8*(L>>4) + r, col L&15
// ---------------------------------------------------------------------------
template<int NCOLS>
__global__ void gat_gemm_wmma(const float* __restrict__ A,
                              const float* __restrict__ B,
                              float* __restrict__ C, int M, int K) {
  const int lane = threadIdx.x & 31;
  const int wave = (int)((blockIdx.x * blockDim.x + threadIdx.x) >> 5);
  const int ntn   = NCOLS / 16;
  const int tiles = (M / 16) * ntn;
  if (wave >= tiles) return;                 // whole-wave uniform exit
  const int tm = wave / ntn;
  const int tn = wave - tm * ntn;
  const int row   = tm * 16 + (lane & 15);
  const int col   = tn * 16 + (lane & 15);
  const int khalf = (lane >> 4) << 1;        // 0 or 2

  v8f acc = {};
  #pragma unroll 4
  for (int k0 = 0; k0 < K; k0 += 4) {
    v2f a = *(const v2f*)(A + (long)row * K + k0 + khalf);   // 8B-aligned b64 load
    const float* bp = B + (long)(k0 + khalf) * NCOLS + col;
    v2f b; b.x = bp[0]; b.y = bp[NCOLS];
    acc = __builtin_amdgcn_wmma_f32_16x16x4_f32(
        /*neg_a=*/false, a, /*neg_b=*/false, b,
        /*c_mod=*/(short)0, acc, /*reuse_a=*/false, /*reuse_b=*/false);
  }

  const int orow = tm * 16 + (lane >> 4) * 8;
  #pragma unroll
  for (int r = 0; r < 8; ++r)
    C[(long)(orow + r) * NCOLS + col] = acc[r];
}

// ---------------------------------------------------------------------------
// a_src[n,h] = sum_c xw[n,h,c]*att_src[h,c];  a_dst likewise. t = n*H + h.
// ---------------------------------------------------------------------------
__global__ void gat_attn_coef(const float* __restrict__ xw,
                              const float* __restrict__ att_s,
                              const float* __restrict__ att_d,
                              float* __restrict__ a_s, float* __restrict__ a_d,
                              int total, int H, int C) {
  int t = blockIdx.x * blockDim.x + threadIdx.x;
  if (t >= total) return;
  int h = t % H;
  const float* xp = xw + (long)t * C;
  float s0 = 0.f, s1 = 0.f;
  for (int c = 0; c < C; ++c) {
    float v = xp[c];
    s0 += v * att_s[h * C + c];
    s1 += v * att_d[h * C + c];
  }
  a_s[t] = s0;
  a_d[t] = s1;
}

// ---------------------------------------------------------------------------
// One wave per edge. Each lane covers V = H*C/32 contiguous channels (all in
// one head). Accumulates numerator sum_w*xw[src] into acc[dst] and denominator
// sum_w into s[dst,h] with native f32 atomics (L2-resident).
// Edge id >= E encodes self-loop (src = dst = id - E).
// ---------------------------------------------------------------------------
template<int H, int C>
__global__ void gat_edge_pass(const int* __restrict__ ei, int E, int Etot,
                              const float* __restrict__ a_s,
                              const float* __restrict__ a_d,
                              const float* __restrict__ xw,
                              float* __restrict__ acc, float* __restrict__ s) {
  constexpr int V = (H * C) / 32;
  const int lane = threadIdx.x & 31;
  const int e = (int)((blockIdx.x * blockDim.x + threadIdx.x) >> 5);
  if (e >= Etot) return;                     // whole-wave uniform exit
  int src, dst;
  if (e < E) { src = ei[e]; dst = ei[E + e]; }
  else       { src = dst = e - E; }

  const int ch = lane * V;
  const int h  = ch / C;
  float z = a_s[src * H + h] + a_d[dst * H + h];
  z = z > 0.f ? z : 0.2f * z;                // LeakyReLU(0.2)
  const float w = __expf(z);

  const float* xp = xw  + (long)src * (H * C) + ch;
  float*       op = acc + (long)dst * (H * C) + ch;
  #pragma unroll
  for (int i = 0; i < V; ++i)
    unsafeAtomicAdd(op + i, w * xp[i]);      // global_atomic_add_f32
  if ((lane & (C / V - 1)) == 0)
    unsafeAtomicAdd(s + dst * H + h, w);
}

// ---------------------------------------------------------------------------
// Layer-1 epilogue: normalize by softmax denom, +bias, BatchNorm(eval), ELU.
// In-place on acc (becomes h for layer 2).
// ---------------------------------------------------------------------------
__global__ void gat_finalize1(float* __restrict__ acc, const float* __restrict__ s,
                              const float* __restrict__ b,
                              const float* __restrict__ gamma,
                              const float* __restrict__ beta,
                              const float* __restrict__ mu,
                              const float* __restrict__ var, int total) {
  int t = blockIdx.x * blockDim.x + threadIdx.x;
  if (t >= total) return;
  int n = t >> 7, j = t & 127, h = j >> 5;
  float v = acc[t] / (s[n * 4 + h] + 1e-16f) + b[j];
  v = (v - mu[j]) * rsqrtf(var[j] + 1e-5f) * gamma[j] + beta[j];
  v = v > 0.f ? v : expm1f(v);               // ELU
  acc[t] = v;
}

// Layer-2 epilogue: normalize + bias (H=1, concat=False is identity on 1 head).
__global__ void gat_finalize2(float* __restrict__ out, const float* __restrict__ s,
                              const float* __restrict__ b, int total) {
  int t = blockIdx.x * blockDim.x + threadIdx.x;
  if (t >= total) return;
  int n = t >> 6, j = t & 63;
  out[t] = out[t] / (s[n] + 1e-16f) + b[j];
}

// ---------------------------------------------------------------------------
extern "C" void kernel_launch(void* const* d_in, const int* in_sizes, int n_in,
                              void* d_out, int out_size, void* d_ws, size_t ws_size,
                              hipStream_t stream) {
  const float* x   = (const float*)d_in[0];
  const int*   ei  = (const int*)  d_in[1];
  const float* W1  = (const float*)d_in[2];
  const float* as1 = (const float*)d_in[3];
  const float* ad1 = (const float*)d_in[4];
  const float* b1  = (const float*)d_in[5];
  const float* g   = (const float*)d_in[6];
  const float* be  = (const float*)d_in[7];
  const float* mu  = (const float*)d_in[8];
  const float* var = (const float*)d_in[9];
  const float* W2  = (const float*)d_in[10];
  const float* as2 = (const float*)d_in[11];
  const float* ad2 = (const float*)d_in[12];
  const float* b2  = (const float*)d_in[13];

  const int N = NODES;
  const int E = in_sizes[1] / 2;
  const int Etot = E + N;

  // Workspace carve-up (floats): ~67 MB total.
  float* ws    = (float*)d_ws;
  float* xw1   = ws;                 // N*128
  float* acc1  = xw1   + (long)N * 128;   // N*128 (becomes h)
  float* asrc1 = acc1  + (long)N * 128;   // N*4
  float* adst1 = asrc1 + N * 4;
  float* s1    = adst1 + N * 4;           // N*4
  float* xw2   = s1    + N * 4;           // N*64
  float* asrc2 = xw2   + (long)N * 64;    // N
  float* adst2 = asrc2 + N;               // N
  float* s2    = adst2 + N;               // N
  float* out   = (float*)d_out;           // N*64, doubles as layer-2 accumulator

  hipMemsetAsync(acc1, 0, (size_t)N * 128 * sizeof(float), stream);
  hipMemsetAsync(s1,   0, (size_t)N * 4   * sizeof(float), stream);
  hipMemsetAsync(out,  0, (size_t)N * 64  * sizeof(float), stream);
  hipMemsetAsync(s2,   0, (size_t)N       * sizeof(float), stream);

  const int BT = 256;                       // 8 waves per block
  const int edgeBlocks = (Etot + 7) / 8;

  // ----- Layer 1: H=4, C=32 -----
  gat_gemm_wmma<128><<<(N / 16) * 8 / 8, BT, 0, stream>>>(x, W1, xw1, N, 128);
  gat_attn_coef<<<(N * 4 + BT - 1) / BT, BT, 0, stream>>>(xw1, as1, ad1,
                                                          asrc1, adst1, N * 4, 4, 32);
  gat_edge_pass<4, 32><<<edgeBlocks, BT, 0, stream>>>(ei, E, Etot, asrc1, adst1,
                                                      xw1, acc1, s1);
  gat_finalize1<<<(N * 128 + BT - 1) / BT, BT, 0, stream>>>(acc1, s1, b1, g, be,
                                                            mu, var, N * 128);

  // ----- Layer 2: H=1, C=64 -----
  gat_gemm_wmma<64><<<((N / 16) * 4 + 7) / 8, BT, 0, stream>>>(acc1, W2, xw2, N, 128);
  gat_attn_coef<<<(N + BT - 1) / BT, BT, 0, stream>>>(xw2, as2, ad2,
                                                      asrc2, adst2, N, 1, 64);
  gat_edge_pass<1, 64><<<edgeBlocks, BT, 0, stream>>>(ei, E, Etot, asrc2, adst2,
                                                      xw2, out, s2);
  gat_finalize2<<<(N * 64 + BT - 1) / BT, BT, 0, stream>>>(out, s2, b2, N * 64);
}